// MeshfreeKANNet_12996571038261
// MI455X (gfx1250) — compile-verified
//
#include <hip/hip_runtime.h>
#include <math.h>

typedef float v2f __attribute__((ext_vector_type(2)));
typedef float v8f __attribute__((ext_vector_type(8)));

#define RADIUS   0.06f
#define INV_R    (1.0f / 0.06f)
#define KAN_GMIN (-1.5f)
#define KAN_H    0.75f
#define KAN_IH   (1.0f / 0.75f)

__device__ __forceinline__ float hatb(float v, int j) {
    float g = KAN_GMIN + KAN_H * (float)j;
    return fmaxf(1.0f - fabsf(v - g) * KAN_IH, 0.0f);
}

__device__ __forceinline__ float softplus_f(float v) {
    // stable: max(x,0) + log1p(exp(-|x|)) == jax.nn.softplus
    return fmaxf(v, 0.0f) + log1pf(expf(-fabsf(v)));
}

__device__ __forceinline__ float cubic_window(float q) {
    // q guaranteed < 1 at call site
    float w_in  = 2.0f / 3.0f + q * q * (4.0f * q - 4.0f);
    float w_out = 4.0f / 3.0f + q * (-4.0f + q * (4.0f - (4.0f / 3.0f) * q));
    return (q <= 0.5f) ? w_in : w_out;
}

// ---------------------------------------------------------------------------
// Phase 1: dense all-pairs pass. One wave per 16-row tile; iterate N in tiles
// of 16 columns. v_wmma_f32_16x16x4_f32 computes the -2*x.n cross term for a
// 16x16 tile in one instruction; VALU epilogue adds norms, applies the cubic
// window and (sparsely) the KAN MLP, accumulating S1 / S2 per row.
// ---------------------------------------------------------------------------
__global__ void __launch_bounds__(256) mkan_phase1(
    const float* __restrict__ x, const float* __restrict__ nodes,
    const float* __restrict__ w, const float* __restrict__ w1a,
    const float* __restrict__ w1b, const float* __restrict__ w2,
    float* __restrict__ S1, float* __restrict__ S2, int M, int N)
{
    __shared__ float sW1a[40], sW1b[40], sW2[40];
    int t = threadIdx.x;
    if (t < 40)       sW1a[t]      = w1a[t];
    else if (t < 80)  sW1b[t - 40] = w1b[t - 40];
    else if (t < 120) sW2[t - 80]  = w2[t - 80];
    __syncthreads();

    const int wave = t >> 5;
    const int lane = t & 31;
    const int col  = lane & 15;   // N index within tile; also A's M index
    const int half = lane >> 4;

    const int rowBase = (blockIdx.x * 8 + wave) * 16;
    if (rowBase >= M) return;     // wave-uniform; EXEC stays all-1s for WMMA

    // A operand (16x4 f32): lanes 0-15 hold K=0,1 = (x0,x1); lanes 16-31 = 0
    float ax0 = x[(rowBase + col) * 2 + 0];
    float ax1 = x[(rowBase + col) * 2 + 1];
    v2f A;
    A.x = half ? 0.0f : ax0;
    A.y = half ? 0.0f : ax1;

    // Per-lane row data for C elements: element v maps to row rowBase+v+8*half
    float xr0[8], xr1[8], xsq[8], s1[8], s2[8];
#pragma unroll
    for (int v = 0; v < 8; ++v) {
        int r = rowBase + v + 8 * half;
        xr0[v] = x[r * 2 + 0];
        xr1[v] = x[r * 2 + 1];
        xsq[v] = xr0[v] * xr0[v] + xr1[v] * xr1[v];
        s1[v] = 0.0f;
        s2[v] = 0.0f;
    }

    const float r2 = RADIUS * RADIUS;
    const int ntiles = N >> 4;
    for (int nt = 0; nt < ntiles; ++nt) {
        int n = nt * 16 + col;
        float n0  = nodes[n * 2 + 0];
        float n1  = nodes[n * 2 + 1];
        float wn  = w[n];
        float nsq = n0 * n0 + n1 * n1;

        // B operand (4x16 f32): lanes 0-15 hold K=0,1 = (-2n0,-2n1); rest 0
        v2f B;
        B.x = half ? 0.0f : (-2.0f * n0);
        B.y = half ? 0.0f : (-2.0f * n1);

        v8f C = {};
        C = __builtin_amdgcn_wmma_f32_16x16x4_f32(
                /*neg_a=*/false, A, /*neg_b=*/false, B,
                /*c_mod=*/(short)0, C, /*reuse_a=*/false, /*reuse_b=*/false);

#pragma unroll
        for (int v = 0; v < 8; ++v) {
            float d2 = C[v] + xsq[v] + nsq;   // |x|^2 - 2 x.n + |n|^2
            d2 = fmaxf(d2, 0.0f);
            if (d2 < r2) {                     // window == 0 outside radius
                float dist = sqrtf(d2);
                float q    = dist * INV_R;
                float win  = cubic_window(q);
                float dx0  = (xr0[v] - n0) * INV_R;
                float dx1  = (xr1[v] - n1) * INV_R;
                float b0[5], b1[5];
#pragma unroll
                for (int j = 0; j < 5; ++j) { b0[j] = hatb(dx0, j); b1[j] = hatb(dx1, j); }
                float acc = 0.0f;
#pragma unroll
                for (int i = 0; i < 8; ++i) {
                    float h = 0.0f;
#pragma unroll
                    for (int j = 0; j < 5; ++j)
                        h += b0[j] * sW1a[i * 5 + j] + b1[j] * sW1b[i * 5 + j];
#pragma unroll
                    for (int j = 0; j < 5; ++j)
                        acc += hatb(h, j) * sW2[i * 5 + j];
                }
                float val = softplus_f(acc) * win;
                s1[v] += val;
                s2[v] += val * wn;
            }
        }
    }

    // Reduce over the 16 lanes of each half (xor masks 1,2,4,8 stay in-half)
#pragma unroll
    for (int v = 0; v < 8; ++v) {
#pragma unroll
        for (int m = 1; m < 16; m <<= 1) {
            s1[v] += __shfl_xor(s1[v], m, 32);
            s2[v] += __shfl_xor(s2[v], m, 32);
        }
    }
    if (col == 0) {
#pragma unroll
        for (int v = 0; v < 8; ++v) {
            int r = rowBase + v + 8 * half;
            S1[r] = s1[v];
            S2[r] = s2[v];
        }
    }
}

// ---------------------------------------------------------------------------
// Phase 2: finalize. Normal rows: S2/(S1+1e-12). Orphan rows (S1 < 1e-14):
// serial kNN scan with register-sorted top-8 and exp(-alpha d) weighting.
// ---------------------------------------------------------------------------
__global__ void __launch_bounds__(256) mkan_phase2(
    const float* __restrict__ x, const float* __restrict__ nodes,
    const float* __restrict__ w, const float* __restrict__ S1,
    const float* __restrict__ S2, float* __restrict__ out, int M, int N)
{
    int m = blockIdx.x * blockDim.x + threadIdx.x;
    if (m >= M) return;
    float s1 = S1[m];
    if (s1 >= 1e-14f) {
        out[m] = S2[m] / (s1 + 1e-12f);
        return;
    }
    float x0 = x[m * 2 + 0], x1 = x[m * 2 + 1];
    float d[8], wv[8];
#pragma unroll
    for (int k = 0; k < 8; ++k) { d[k] = 3.0e38f; wv[k] = 0.0f; }
    for (int n = 0; n < N; ++n) {
        float dx = x0 - nodes[n * 2 + 0];
        float dy = x1 - nodes[n * 2 + 1];
        float dist = sqrtf(dx * dx + dy * dy);
        if (dist < d[7]) {
            float cd = dist, cw = w[n];
#pragma unroll
            for (int k = 0; k < 8; ++k) {
                if (cd < d[k]) {
                    float td = d[k], tw = wv[k];
                    d[k] = cd;  wv[k] = cw;
                    cd = td;    cw = tw;
                }
            }
        }
    }
    const float alpha = 20.0f / RADIUS;
    float s = 0.0f, sw = 0.0f;
#pragma unroll
    for (int k = 0; k < 8; ++k) {
        float e = expf(-alpha * d[k]);
        s  += e;
        sw += e * wv[k];
    }
    out[m] = sw / (s + 1e-18f);
}

extern "C" void kernel_launch(void* const* d_in, const int* in_sizes, int n_in,
                              void* d_out, int out_size, void* d_ws, size_t ws_size,
                              hipStream_t stream)
{
    const float* x     = (const float*)d_in[0];
    const float* nodes = (const float*)d_in[1];
    const float* w     = (const float*)d_in[2];
    const float* w1a   = (const float*)d_in[3];
    const float* w1b   = (const float*)d_in[4];
    const float* w2    = (const float*)d_in[5];
    float* out = (float*)d_out;

    int M = in_sizes[0] / 2;
    int N = in_sizes[1] / 2;

    float* S1 = (float*)d_ws;
    float* S2 = S1 + M;

    int rowTiles = (M + 15) / 16;
    int blocks1  = (rowTiles + 7) / 8;          // 8 waves (row tiles) per block
    mkan_phase1<<<blocks1, 256, 0, stream>>>(x, nodes, w, w1a, w1b, w2, S1, S2, M, N);

    int blocks2 = (M + 255) / 256;
    mkan_phase2<<<blocks2, 256, 0, stream>>>(x, nodes, w, S1, S2, out, M, N);
}